// ImprovedPressureGNN_27762668601577
// MI455X (gfx1250) — compile-verified
//
#include <hip/hip_runtime.h>
#include <hip/hip_bf16.h>

typedef __attribute__((ext_vector_type(2))) float v2f;
typedef __attribute__((ext_vector_type(8))) float v8f;

#define HDIM 64

// ---------------------------------------------------------------------------
// Degree / normalization kernels
// ---------------------------------------------------------------------------
__global__ void deg_init_kernel(float* __restrict__ deg, int N) {
    int i = blockIdx.x * blockDim.x + threadIdx.x;
    if (i < N) deg[i] = 1.0f;  // self-loop contributes 1
}

__global__ void deg_accum_kernel(const long long* __restrict__ col,
                                 float* __restrict__ deg, int E) {
    int e = blockIdx.x * blockDim.x + threadIdx.x;
    if (e < E) atomicAdd(&deg[(int)col[e]], 1.0f);
}

__global__ void deg_rsqrt_kernel(float* __restrict__ deg, int N) {
    int i = blockIdx.x * blockDim.x + threadIdx.x;
    if (i < N) {
        float d = deg[i];
        deg[i] = d > 0.0f ? rsqrtf(d) : 0.0f;
    }
}

// ---------------------------------------------------------------------------
// WMMA fp32 GEMM: out[N x 64] = A[N x K] @ B[K x 64] (+bias) (+relu)
// One block = 256 threads = 8 waves; each wave owns a 16-row stripe and
// produces 16x64 via 4 accumulators of V_WMMA_F32_16X16X4_F32.
// ---------------------------------------------------------------------------
__global__ void gemm_wmma_kernel(const float* __restrict__ A,
                                 const float* __restrict__ B,
                                 const float* __restrict__ bias,
                                 float* __restrict__ out,
                                 int N, int K, int do_relu) {
    __shared__ float sB[64 * HDIM];  // up to K=64

    int tid = threadIdx.x;
    for (int i = tid; i < K * HDIM; i += blockDim.x) sB[i] = B[i];
    __syncthreads();

    int wave = tid >> 5;
    int lane = tid & 31;
    int rowBase = blockIdx.x * 128 + wave * 16;
    if (rowBase >= N) return;  // uniform per-wave exit; EXEC stays all-1 for WMMA waves

    int m     = lane & 15;         // M (for A) / N-column (for B)
    int khalf = (lane >> 4) * 2;   // lanes 0-15: K=0,1 ; lanes 16-31: K=2,3
    int row   = rowBase + m;
    int rowc  = row < N ? row : (N - 1);  // clamp loads; stores are predicated

    v8f acc[4] = {v8f{}, v8f{}, v8f{}, v8f{}};

    for (int k0 = 0; k0 < K; k0 += 4) {
        v2f a;
        a.x = A[(long long)rowc * K + k0 + khalf];
        a.y = A[(long long)rowc * K + k0 + khalf + 1];
#pragma unroll
        for (int ct = 0; ct < 4; ++ct) {
            int col = ct * 16 + m;
            v2f b;
            b.x = sB[(k0 + khalf) * HDIM + col];
            b.y = sB[(k0 + khalf + 1) * HDIM + col];
            acc[ct] = __builtin_amdgcn_wmma_f32_16x16x4_f32(
                false, a, false, b, (short)0, acc[ct], false, false);
        }
    }

    // C/D layout: VGPR r -> lanes 0-15 hold M=r, lanes 16-31 hold M=r+8
    int mBase = rowBase + (lane >> 4) * 8;
#pragma unroll
    for (int ct = 0; ct < 4; ++ct) {
        int col = ct * 16 + m;
        float bv = bias ? bias[col] : 0.0f;
#pragma unroll
        for (int r = 0; r < 8; ++r) {
            int orow = mBase + r;
            if (orow < N) {
                float v = acc[ct][r] + bv;
                if (do_relu) v = fmaxf(v, 0.0f);
                out[(long long)orow * HDIM + col] = v;
            }
        }
    }
}

// ---------------------------------------------------------------------------
// agg init: agg[n, f] = b_conv[f]
// ---------------------------------------------------------------------------
__global__ void agg_init_kernel(float* __restrict__ agg,
                                const float* __restrict__ bias, long long total) {
    long long i = (long long)blockIdx.x * blockDim.x + threadIdx.x;
    if (i < total) agg[i] = bias[(int)(i & (HDIM - 1))];
}

// ---------------------------------------------------------------------------
// Scatter: agg[col] += t[row] * dinv[row]*dinv[col]; edges E..E+N-1 are
// self-loops. 32 threads per edge, 2 features per thread.
// ---------------------------------------------------------------------------
__global__ void scatter_add_kernel(const long long* __restrict__ rowIdx,
                                   const long long* __restrict__ colIdx,
                                   const float* __restrict__ dinv,
                                   const float* __restrict__ t,
                                   float* __restrict__ agg,
                                   int E, int N) {
    long long gid = (long long)blockIdx.x * blockDim.x + threadIdx.x;
    long long tot = (long long)(E + N) * 32;
    if (gid >= tot) return;
    int e = (int)(gid >> 5);
    int f = ((int)gid & 31) * 2;
    int r, c;
    if (e < E) {
        r = (int)rowIdx[e];
        c = (int)colIdx[e];
    } else {
        r = c = e - E;
    }
    float nrm = dinv[r] * dinv[c];
    const float* tp = t + (long long)r * HDIM + f;
    float* ap = agg + (long long)c * HDIM + f;
    atomicAdd(ap, tp[0] * nrm);
    atomicAdd(ap + 1, tp[1] * nrm);
}

// ---------------------------------------------------------------------------
// LayerNorm -> relu -> +residual. One wave32 per node, 2 features per lane.
// ---------------------------------------------------------------------------
__global__ void ln_relu_res_kernel(const float* __restrict__ agg,
                                   const float* __restrict__ hprev,
                                   const float* __restrict__ gamma,
                                   const float* __restrict__ beta,
                                   float* __restrict__ hout, int N) {
    long long gt = (long long)blockIdx.x * blockDim.x + threadIdx.x;
    int node = (int)(gt >> 5);
    int lane = threadIdx.x & 31;
    if (node >= N) return;
    const float* a = agg + (long long)node * HDIM;
    float x0 = a[lane];
    float x1 = a[lane + 32];

    float s = x0 + x1;
#pragma unroll
    for (int off = 16; off > 0; off >>= 1) s += __shfl_xor(s, off, 32);
    float mu = s * (1.0f / HDIM);

    float d0 = x0 - mu, d1 = x1 - mu;
    float v = d0 * d0 + d1 * d1;
#pragma unroll
    for (int off = 16; off > 0; off >>= 1) v += __shfl_xor(v, off, 32);
    float inv = rsqrtf(v * (1.0f / HDIM) + 1e-5f);

    const float* hp = hprev + (long long)node * HDIM;
    float y0 = fmaxf(d0 * inv * gamma[lane] + beta[lane], 0.0f) + hp[lane];
    float y1 = fmaxf(d1 * inv * gamma[lane + 32] + beta[lane + 32], 0.0f) + hp[lane + 32];
    float* ho = hout + (long long)node * HDIM;
    ho[lane] = y0;
    ho[lane + 32] = y1;
}

// ---------------------------------------------------------------------------
// Output projection: out[n] = dot(h[n], W_out) + b_out. One wave32 per node.
// ---------------------------------------------------------------------------
__global__ void out_proj_kernel(const float* __restrict__ h,
                                const float* __restrict__ W_out,
                                const float* __restrict__ b_out,
                                float* __restrict__ out, int N) {
    long long gt = (long long)blockIdx.x * blockDim.x + threadIdx.x;
    int node = (int)(gt >> 5);
    int lane = threadIdx.x & 31;
    if (node >= N) return;
    const float* hp = h + (long long)node * HDIM;
    float p = hp[lane] * W_out[lane] + hp[lane + 32] * W_out[lane + 32];
#pragma unroll
    for (int off = 16; off > 0; off >>= 1) p += __shfl_xor(p, off, 32);
    if (lane == 0) out[node] = p + b_out[0];
}

// ---------------------------------------------------------------------------
// Host-side launch
// ---------------------------------------------------------------------------
extern "C" void kernel_launch(void* const* d_in, const int* in_sizes, int n_in,
                              void* d_out, int out_size, void* d_ws, size_t ws_size,
                              hipStream_t stream) {
    const float*     x      = (const float*)d_in[0];
    const long long* ei     = (const long long*)d_in[1];  // [2, E] int64
    const float*     W_in   = (const float*)d_in[2];
    const float*     b_in   = (const float*)d_in[3];
    const float*     W_conv = (const float*)d_in[4];
    const float*     b_conv = (const float*)d_in[5];
    const float*     gamma  = (const float*)d_in[6];
    const float*     beta   = (const float*)d_in[7];
    const float*     W_out  = (const float*)d_in[8];
    const float*     b_out  = (const float*)d_in[9];
    float*           outp   = (float*)d_out;

    const int IN = 16;
    const int N  = in_sizes[0] / IN;
    const int E  = in_sizes[1] / 2;
    const int L  = in_sizes[4] / (HDIM * HDIM);

    // Workspace carve-out
    char* ws = (char*)d_ws;
    size_t off = 0;
    auto carve = [&](size_t bytes) {
        void* p = ws + off;
        off += (bytes + 255) & ~(size_t)255;
        return p;
    };
    float* dinv = (float*)carve((size_t)N * sizeof(float));
    float* bufA = (float*)carve((size_t)N * HDIM * sizeof(float));  // h
    float* bufB = (float*)carve((size_t)N * HDIM * sizeof(float));  // t
    float* bufC = (float*)carve((size_t)N * HDIM * sizeof(float));  // agg
    (void)ws_size; (void)n_in; (void)out_size;

    const long long* rowIdx = ei;
    const long long* colIdx = ei + E;

    const int TB = 256;

    // 1) degree -> dinv
    deg_init_kernel<<<(N + TB - 1) / TB, TB, 0, stream>>>(dinv, N);
    deg_accum_kernel<<<(E + TB - 1) / TB, TB, 0, stream>>>(colIdx, dinv, E);
    deg_rsqrt_kernel<<<(N + TB - 1) / TB, TB, 0, stream>>>(dinv, N);

    // 2) input layer: h = relu(x @ W_in + b_in)
    int gemmBlocks = (N + 127) / 128;
    gemm_wmma_kernel<<<gemmBlocks, TB, 0, stream>>>(x, W_in, b_in, bufA, N, IN, 1);

    float* h   = bufA;
    float* t   = bufB;
    float* agg = bufC;

    long long totNH = (long long)N * HDIM;
    long long scatterThreads = (long long)(E + N) * 32;
    int scatterBlocks = (int)((scatterThreads + TB - 1) / TB);
    int waveNodeBlocks = (int)(((long long)N * 32 + TB - 1) / TB);

    for (int l = 0; l < L; ++l) {
        // t = h @ W_conv[l]
        gemm_wmma_kernel<<<gemmBlocks, TB, 0, stream>>>(
            h, W_conv + (size_t)l * HDIM * HDIM, nullptr, t, N, HDIM, 0);
        // agg = b_conv[l]
        agg_init_kernel<<<(int)((totNH + TB - 1) / TB), TB, 0, stream>>>(
            agg, b_conv + (size_t)l * HDIM, totNH);
        // agg[col] += t[row] * norm  (incl. self-loops)
        scatter_add_kernel<<<scatterBlocks, TB, 0, stream>>>(
            rowIdx, colIdx, dinv, t, agg, E, N);
        // h_new = relu(LN(agg)) + h ; write into t buffer, then swap
        ln_relu_res_kernel<<<waveNodeBlocks, TB, 0, stream>>>(
            agg, h, gamma + (size_t)l * HDIM, beta + (size_t)l * HDIM, t, N);
        float* tmp = h; h = t; t = tmp;
    }

    // 3) out = h @ W_out + b_out
    out_proj_kernel<<<waveNodeBlocks, TB, 0, stream>>>(h, W_out, b_out, outp, N);
}